// MixtureOfExperts_3633542332636
// MI455X (gfx1250) — compile-verified
//
#include <hip/hip_runtime.h>
#include <math.h>

// MoE forward for MI455X (gfx1250, wave32).
// Per 16-token tile: 7 WMMA (fc1) + 1 WMMA (gate) + 25 WMMA (fc2) of
// v_wmma_f32_16x16x4_f32, fc1/gate with inline-0 C, fc2 C = b2 (b1/bg folded into the
// VALU epilogue to keep VGPR count < 256).  h is transposed through
// wave-private LDS ([H][m], stride 20 dwords -> conflict-free b128
// stores / b32 loads).  Softmax + gating fused via an LDS broadcast
// of exp(); no block barriers (same-wave DS is in-order).

typedef __attribute__((ext_vector_type(2))) float v2f;
typedef __attribute__((ext_vector_type(4))) float v4f;
typedef __attribute__((ext_vector_type(8))) float v8f;

#define IN_DIM   3
#define HID      20
#define OUT_DIM  3
#define NEXP     5
#define NHID     100              // NEXP*HID
#define NTILE_H  7                // ceil(NHID/16)
#define LDS_STRIDE 20             // dwords per H-row (16 tokens + pad, 16B-aligned, bank-perfect)
#define LDS_HROWS  112            // 7 tiles * 16 columns (incl. zero-padded cols 100..111)
#define LDS_PER_WAVE (LDS_HROWS*LDS_STRIDE + NEXP*16)   // h buffer + exp broadcast buffer
#define WAVES_PER_BLOCK 4

static __device__ __forceinline__ v8f splat8(float s) {
  v8f v;
#pragma unroll
  for (int i = 0; i < 8; ++i) v[i] = s;
  return v;
}

__global__ __launch_bounds__(WAVES_PER_BLOCK * 32) void moe_wmma_kernel(
    const float* __restrict__ x,  const float* __restrict__ W1, const float* __restrict__ b1,
    const float* __restrict__ W2, const float* __restrict__ b2,
    const float* __restrict__ Wg, const float* __restrict__ bg,
    float* __restrict__ out_mixed, float* __restrict__ out_gate, int nTiles)
{
  __shared__ __align__(16) float lds[WAVES_PER_BLOCK * LDS_PER_WAVE];
  const int lane = threadIdx.x & 31;
  const int wv   = threadIdx.x >> 5;
  float* hbuf = lds + wv * LDS_PER_WAVE;
  float* ebuf = hbuf + LDS_HROWS * LDS_STRIDE;

  const int n  = lane & 15;   // column index within 16-wide tile
  const int hi = lane >> 4;   // half-wave select (rows/K split)

  // ---------------- loop-invariant weights in registers ----------------
  // fc1 B-tiles: B is [K=4 x N=16]; v0 = row k=2*hi, v1 = row k=2*hi+1 (k=3 row = 0 pad)
  v2f w1B[NTILE_H]; float b1l[NTILE_H];
#pragma unroll
  for (int t = 0; t < NTILE_H; ++t) {
    int H = t * 16 + n;
    v2f b; b.x = 0.f; b.y = 0.f; float bb = 0.f;
    if (H < NHID) {
      int e = H / HID, j = H - e * HID;
      b.x = W1[e * (IN_DIM * HID) + (2 * hi) * HID + j];
      b.y = hi ? 0.f : W1[e * (IN_DIM * HID) + HID + j];
      bb  = b1[H];
    }
    w1B[t] = b; b1l[t] = bb;
  }
  // gate B-tile (Wg is [IN_DIM][NEXP])
  v2f wgB; wgB.x = 0.f; wgB.y = 0.f; float bgl = 0.f;
  if (n < NEXP) {
    wgB.x = Wg[(2 * hi) * NEXP + n];
    wgB.y = hi ? 0.f : Wg[NEXP + n];
    bgl   = bg[n];
  }
  // fc2 B-tiles per expert: 5 K-chunks of 4, N cols = OUT_DIM (rest 0)
  v2f w2B[NEXP][5]; float b2l[NEXP];
#pragma unroll
  for (int e = 0; e < NEXP; ++e) {
#pragma unroll
    for (int kt = 0; kt < 5; ++kt) {
      int j = kt * 4 + 2 * hi;
      v2f b; b.x = 0.f; b.y = 0.f;
      if (n < OUT_DIM) {
        b.x = W2[e * (HID * OUT_DIM) + j * OUT_DIM + n];
        b.y = W2[e * (HID * OUT_DIM) + (j + 1) * OUT_DIM + n];
      }
      w2B[e][kt] = b;
    }
    b2l[e] = (n < OUT_DIM) ? b2[e * OUT_DIM + n] : 0.f;
  }

  const int waveGlobal = blockIdx.x * WAVES_PER_BLOCK + wv;
  const int nWaves     = gridDim.x * WAVES_PER_BLOCK;

  for (int tile = waveGlobal; tile < nTiles; tile += nWaves) {
    const int base = tile * 16;

    // ---- load x as A-tile, branch-free (K=3 row of every B-tile is 0,
    //      so the K=3 slot of A may hold arbitrary finite junk) ----
    const float* xr = x + (size_t)(base + n) * IN_DIM;
    v2f a;
    a.x = xr[2 * hi];   // hi=0: x[m][0], hi=1: x[m][2]
    a.y = xr[1];        // hi=0: x[m][1], hi=1: junk * 0-row

    // ---- fc1: 7x WMMA (C = inline 0), bias+relu in epilogue, transpose-store ----
#pragma unroll
    for (int t = 0; t < NTILE_H; ++t) {
      v8f c = {};
      c = __builtin_amdgcn_wmma_f32_16x16x4_f32(false, a, false, w1B[t],
                                                (short)0, c, false, false);
      v4f lo, hiv;
#pragma unroll
      for (int r = 0; r < 4; ++r) {
        lo[r]  = fmaxf(c[r]     + b1l[t], 0.f);
        hiv[r] = fmaxf(c[r + 4] + b1l[t], 0.f);
      }
      // lane owns column H = 16t+n, rows m = 8*hi .. 8*hi+7 (contiguous) -> b128 x2
      float* p = hbuf + (t * 16 + n) * LDS_STRIDE + 8 * hi;
      *(v4f*)p       = lo;
      *(v4f*)(p + 4) = hiv;
    }

    // ---- gate: 1x WMMA (C = inline 0), bias in epilogue, exp(), LDS broadcast ----
    v4f eLo, eHi;
    {
      v8f c = {};
      c = __builtin_amdgcn_wmma_f32_16x16x4_f32(false, a, false, wgB,
                                                (short)0, c, false, false);
#pragma unroll
      for (int r = 0; r < 4; ++r) {
        eLo[r] = __expf(c[r]     + bgl);
        eHi[r] = __expf(c[r + 4] + bgl);
      }
      if (n < NEXP) {
        float* p = ebuf + n * 16 + 8 * hi;   // [e][m] layout, m-contiguous
        *(v4f*)p       = eLo;
        *(v4f*)(p + 4) = eHi;
      }
    }

    // same-wave DS is in-order; just stop the compiler from reordering
    asm volatile("" ::: "memory");

    // ---- fc2: 25x WMMA, per-expert accumulators (b2 folded into C-init) ----
    v8f S[NEXP];
#pragma unroll
    for (int e = 0; e < NEXP; ++e) {
      S[e] = splat8(b2l[e]);
#pragma unroll
      for (int kt = 0; kt < 5; ++kt) {
        int H0 = e * HID + kt * 4 + 2 * hi;           // A-tile: row m=n, K rows H0,H0+1
        const float* p = hbuf + H0 * LDS_STRIDE + n;
        v2f a2; a2.x = p[0]; a2.y = p[LDS_STRIDE];
        S[e] = __builtin_amdgcn_wmma_f32_16x16x4_f32(false, a2, false, w2B[e][kt],
                                                     (short)0, S[e], false, false);
      }
    }

    // ---- epilogue: softmax denom + gated mix + outputs ----
    v4f exl[NEXP], exh[NEXP];
#pragma unroll
    for (int e = 0; e < NEXP; ++e) {                  // broadcast reads (same addr within each half-wave)
      const float* p = ebuf + e * 16 + 8 * hi;
      exl[e] = *(const v4f*)p;
      exh[e] = *(const v4f*)(p + 4);
    }
#pragma unroll
    for (int r = 0; r < 8; ++r) {
      float ee[NEXP];
#pragma unroll
      for (int e = 0; e < NEXP; ++e) ee[e] = (r < 4) ? exl[e][r] : exh[e][r - 4];
      float d  = ((ee[0] + ee[1]) + (ee[2] + ee[3])) + ee[4];
      float rd = __builtin_amdgcn_rcpf(d);
      float acc = 0.f;
#pragma unroll
      for (int e = 0; e < NEXP; ++e) acc = fmaf(ee[e], S[e][r], acc);
      int m = base + r + 8 * hi;
      if (n < OUT_DIM) out_mixed[(size_t)m * OUT_DIM + n] = acc * rd;
      float exOwn = (r < 4) ? eLo[r] : eHi[r - 4];
      if (n < NEXP)   out_gate[(size_t)m * NEXP + n] = exOwn * rd;
    }

    asm volatile("" ::: "memory");   // keep next tile's LDS stores after this tile's loads
  }
}

extern "C" void kernel_launch(void* const* d_in, const int* in_sizes, int n_in,
                              void* d_out, int out_size, void* d_ws, size_t ws_size,
                              hipStream_t stream) {
  const float* x  = (const float*)d_in[0];
  const float* W1 = (const float*)d_in[1];
  const float* b1 = (const float*)d_in[2];
  const float* W2 = (const float*)d_in[3];
  const float* b2 = (const float*)d_in[4];
  const float* Wg = (const float*)d_in[5];
  const float* bg = (const float*)d_in[6];

  const int batch  = in_sizes[0] / IN_DIM;   // 1<<20
  const int nTiles = batch / 16;             // batch is a multiple of 16

  float* out_mixed = (float*)d_out;                       // [batch,3]
  float* out_gate  = (float*)d_out + (size_t)batch * 3;   // [batch,5]

  dim3 grid(2048), block(WAVES_PER_BLOCK * 32);
  hipLaunchKernelGGL(moe_wmma_kernel, grid, block, 0, stream,
                     x, W1, b1, W2, b2, Wg, bg, out_mixed, out_gate, nTiles);
}